// MSDeformAttn_214748365280
// MI455X (gfx1250) — compile-verified
//
#include <hip/hip_runtime.h>
#include <hip/hip_bf16.h>
#include <math.h>

// ---------------------------------------------------------------------------
// MS-Deformable-Attention 3D for gfx1250 (MI455X).
//   BS=4, LQ=4096, DM=256, NH=8, NL=4, NP=4, head_dim c=32
//   pyramid: 32^3, 16^3, 8^3, 4^3  -> Len_in = 37440
// ---------------------------------------------------------------------------

#define BS   4
#define LQ   4096
#define DM   256
#define NH   8
#define NL   4
#define NP   4
#define HC   32              // DM / NH
#define LEN_IN 37440

__device__ __constant__ int kLvlSize[NL]  = {32, 16, 8, 4};
__device__ __constant__ int kLvlStart[NL] = {0, 32768, 36864, 37376};

typedef float v2f __attribute__((ext_vector_type(2)));
typedef float v4f __attribute__((ext_vector_type(4)));
typedef float v8f __attribute__((ext_vector_type(8)));

// ---- CDNA5 async global->LDS copy (ASYNCcnt path, ISA 15.18.3 op 98) -------
__device__ __forceinline__ void async_copy_b128(unsigned int lds_byte_off,
                                                const void* gaddr) {
    asm volatile("global_load_async_to_lds_b128 %0, %1, off"
                 :: "v"(lds_byte_off), "v"(gaddr) : "memory");
}
__device__ __forceinline__ void wait_async0() {
    asm volatile("s_wait_asynccnt 0x0" ::: "memory");
}
// low 32 bits of a generic pointer to LDS == workgroup-relative LDS byte offset
__device__ __forceinline__ unsigned int lds_off(const void* p) {
    return (unsigned int)(unsigned long long)p;
}

// ---------------------------------------------------------------------------
// fp32 GEMM with V_WMMA_F32_16X16X4_F32, double-buffered async LDS staging.
//   C[M,N] = A[M,K] @ B[K,N] + bias[N]
// Block = 256 threads (8 waves). Block tile: 64(M) x 128(N), K-slab = 16.
// Waves arranged 4(M) x 2(N); each wave: 16x64 via 4 accumulators of 16x16.
// Requires: M % 64 == 0, N % 128 == 0, K % 16 == 0 (true for all our GEMMs).
// ---------------------------------------------------------------------------
#define BM 64
#define BN 128
#define BK 16

__global__ __launch_bounds__(256)
void gemm_f32_wmma(const float* __restrict__ A, const float* __restrict__ B,
                   const float* __restrict__ bias, float* __restrict__ C,
                   int M, int N, int K)
{
    __shared__ float As[2][BM * BK];   // [m][k]  64 x 16
    __shared__ float Bs[2][BK * BN];   // [k][n]  16 x 128

    const int tid   = threadIdx.x;
    const int lane  = tid & 31;
    const int wave  = tid >> 5;
    const int l16   = lane & 15;
    const int lhalf = lane >> 4;           // 0: lanes 0-15, 1: lanes 16-31

    const int bm = blockIdx.x * BM;
    const int bn = blockIdx.y * BN;
    const int wm = (wave & 3) * 16;        // wave M offset inside block tile
    const int wn = (wave >> 2) * 64;       // wave N offset inside block tile

    // per-thread staging assignment (16B async chunks)
    const int arow = tid >> 2;             // 0..63
    const int acol = (tid & 3) * 4;        // 0,4,8,12
    const int brow = tid >> 5;             // 0..7
    const int bcol = (tid & 31) * 4;       // 0..124

    v8f acc[4] = {v8f{0}, v8f{0}, v8f{0}, v8f{0}};

    // issue one K-slab of async copies into buffer `buf`
    auto stage = [&](int k0, int buf) {
        async_copy_b128(lds_off(&As[buf][arow * BK + acol]),
                        A + (size_t)(bm + arow) * K + (k0 + acol));
        async_copy_b128(lds_off(&Bs[buf][brow * BN + bcol]),
                        B + (size_t)(k0 + brow) * N + (bn + bcol));
        async_copy_b128(lds_off(&Bs[buf][(brow + 8) * BN + bcol]),
                        B + (size_t)(k0 + brow + 8) * N + (bn + bcol));
    };

    stage(0, 0);
    wait_async0();
    __syncthreads();

    int cur = 0;
    for (int k0 = 0; k0 < K; k0 += BK) {
        // prefetch next slab into the other buffer while we compute
        if (k0 + BK < K) stage(k0 + BK, cur ^ 1);

        const float* as = As[cur];
        const float* bs = Bs[cur];
        #pragma unroll
        for (int kk = 0; kk < BK; kk += 4) {
            // A fragment (16x4 fp32): lanes 0-15 hold K=kk,kk+1; lanes 16-31 K=kk+2,kk+3
            v2f afrag;
            afrag.x = as[(wm + l16) * BK + kk + 2 * lhalf + 0];
            afrag.y = as[(wm + l16) * BK + kk + 2 * lhalf + 1];
            #pragma unroll
            for (int t = 0; t < 4; ++t) {
                // B fragment (4x16 fp32): VGPR j holds row kk+j (lanes 0-15) / kk+j+2 (16-31)
                const int ncol = wn + t * 16 + l16;
                v2f bfrag;
                bfrag.x = bs[(kk + 2 * lhalf + 0) * BN + ncol];
                bfrag.y = bs[(kk + 2 * lhalf + 1) * BN + ncol];
                acc[t] = __builtin_amdgcn_wmma_f32_16x16x4_f32(
                    /*neg_a=*/false, afrag, /*neg_b=*/false, bfrag,
                    /*c_mod=*/(short)0, acc[t], /*reuse_a=*/false, /*reuse_b=*/false);
            }
        }

        wait_async0();      // next slab resident in LDS
        __syncthreads();    // visible to all waves; all done reading `cur`
        cur ^= 1;
    }

    // ---- epilogue: C/D layout: VGPR r -> row r (lanes 0-15) / r+8 (lanes 16-31) ----
    #pragma unroll
    for (int t = 0; t < 4; ++t) {
        const int ncol = bn + wn + t * 16 + l16;
        const float bv = bias[ncol];
        #pragma unroll
        for (int r = 0; r < 8; ++r) {
            const int row = bm + wm + r + 8 * lhalf;
            C[(size_t)row * N + ncol] = acc[t][r] + bv;
        }
    }
}

// ---------------------------------------------------------------------------
// Fused softmax + trilinear deformable sampling.
// Grid: BS*LQ blocks of 256 threads; thread = (head h = tid/32, channel = tid%32).
// ---------------------------------------------------------------------------
__global__ __launch_bounds__(256)
void msda3d_sample(const float* __restrict__ value,     // (BS, LEN_IN, DM)
                   const float* __restrict__ off_raw,   // (BS*LQ, 384)
                   const float* __restrict__ attn_raw,  // (BS*LQ, 128)
                   const float* __restrict__ refpts,    // (BS, LQ, NL, 3)
                   float* __restrict__ out_pre)         // (BS*LQ, DM)
{
    const int bq = blockIdx.x;          // 0 .. BS*LQ-1
    const int b  = bq >> 12;            // LQ = 4096
    const int h  = threadIdx.x >> 5;
    const int ch = threadIdx.x & 31;

    const float* off = off_raw  + (size_t)bq * (NH * NL * NP * 3) + h * (NL * NP * 3);
    const float* al  = attn_raw + (size_t)bq * (NH * NL * NP)     + h * (NL * NP);
    const float* ref = refpts   + (size_t)bq * (NL * 3);

    // softmax statistics over the 16 (level, point) logits of this head
    float mx = -INFINITY;
    #pragma unroll
    for (int i = 0; i < NL * NP; ++i) mx = fmaxf(mx, al[i]);
    float denom = 0.f;
    #pragma unroll
    for (int i = 0; i < NL * NP; ++i) denom += __expf(al[i] - mx);
    const float inv_denom = 1.f / denom;

    const float* vbase = value + (size_t)b * LEN_IN * DM + h * HC + ch;

    float acc = 0.f;
    #pragma unroll
    for (int l = 0; l < NL; ++l) {
        const int   D  = kLvlSize[l];          // cube: D == H == W
        const float fD = (float)D;
        const float* vlev = vbase + (size_t)kLvlStart[l] * DM;
        const float r0 = ref[l * 3 + 0];
        const float r1 = ref[l * 3 + 1];
        const float r2 = ref[l * 3 + 2];
        #pragma unroll
        for (int p = 0; p < NP; ++p) {
            const float* o = off + (l * NP + p) * 3;
            const float aw = __expf(al[l * NP + p] - mx) * inv_denom;
            // grid_sample-style (align_corners=False): pixel = loc * size - 0.5
            const float dz = (r0 + o[0] / fD) * fD - 0.5f;
            const float dy = (r1 + o[1] / fD) * fD - 0.5f;
            const float dx = (r2 + o[2] / fD) * fD - 0.5f;
            const float z0f = floorf(dz), y0f = floorf(dy), x0f = floorf(dx);
            const float fz = dz - z0f, fy = dy - y0f, fx = dx - x0f;
            const int z0 = (int)z0f, y0 = (int)y0f, x0 = (int)x0f;
            #pragma unroll
            for (int cz = 0; cz < 2; ++cz) {
                const int   zz = z0 + cz;
                const float wz = cz ? fz : 1.f - fz;
                const int   zc = min(max(zz, 0), D - 1);
                const bool  vz = (zz >= 0) & (zz < D);
                #pragma unroll
                for (int cy = 0; cy < 2; ++cy) {
                    const int   yy = y0 + cy;
                    const float wy = cy ? fy : 1.f - fy;
                    const int   yc = min(max(yy, 0), D - 1);
                    const bool  vy = (yy >= 0) & (yy < D);
                    #pragma unroll
                    for (int cx = 0; cx < 2; ++cx) {
                        const int   xx = x0 + cx;
                        const float wx = cx ? fx : 1.f - fx;
                        const int   xc = min(max(xx, 0), D - 1);
                        const bool  valid = vz & vy & ((xx >= 0) & (xx < D));
                        const int   idx = (zc * D + yc) * D + xc;
                        const float g = vlev[(size_t)idx * DM];  // coalesced over 32 channels
                        acc += valid ? (aw * wz * wy * wx) * g : 0.f;
                    }
                }
            }
        }
    }
    out_pre[(size_t)bq * DM + h * HC + ch] = acc;
}

// ---------------------------------------------------------------------------
// Launcher. setup_inputs() order:
//  0 query  1 reference_points  2 input_flatten  3 spatial_shapes  4 level_start
//  5 W_off  6 b_off  7 W_attn  8 b_attn  9 W_val 10 b_val 11 W_out 12 b_out
// ---------------------------------------------------------------------------
extern "C" void kernel_launch(void* const* d_in, const int* in_sizes, int n_in,
                              void* d_out, int out_size, void* d_ws, size_t ws_size,
                              hipStream_t stream)
{
    const float* query  = (const float*)d_in[0];
    const float* refp   = (const float*)d_in[1];
    const float* inflat = (const float*)d_in[2];
    const float* W_off  = (const float*)d_in[5];
    const float* b_off  = (const float*)d_in[6];
    const float* W_attn = (const float*)d_in[7];
    const float* b_attn = (const float*)d_in[8];
    const float* W_val  = (const float*)d_in[9];
    const float* b_val  = (const float*)d_in[10];
    const float* W_out  = (const float*)d_in[11];
    const float* b_out  = (const float*)d_in[12];
    float* out = (float*)d_out;

    const int M_val  = BS * LEN_IN;        // 149760  (% 64 == 0)
    const int M_q    = BS * LQ;            // 16384
    const int N_off  = NH * NL * NP * 3;   // 384
    const int N_attn = NH * NL * NP;       // 128

    // workspace layout
    char* ws = (char*)d_ws;
    float* ws_value = (float*)(ws);                                   // 149760*256 f
    float* ws_off   = (float*)(ws + (size_t)M_val * DM * 4);          // 16384*384 f
    float* ws_attn  = (float*)((char*)ws_off + (size_t)M_q * N_off * 4);
    float* ws_outp  = (float*)((char*)ws_attn + (size_t)M_q * N_attn * 4);

    // 1) value projection: (BS*LEN_IN,256) @ (256,256) + b_val
    gemm_f32_wmma<<<dim3(M_val / BM, DM / BN), 256, 0, stream>>>(
        inflat, W_val, b_val, ws_value, M_val, DM, DM);

    // 2) sampling-offset logits: (16384,256) @ (256,384) + b_off
    gemm_f32_wmma<<<dim3(M_q / BM, N_off / BN), 256, 0, stream>>>(
        query, W_off, b_off, ws_off, M_q, N_off, DM);

    // 3) attention logits: (16384,256) @ (256,128) + b_attn
    gemm_f32_wmma<<<dim3(M_q / BM, N_attn / BN), 256, 0, stream>>>(
        query, W_attn, b_attn, ws_attn, M_q, N_attn, DM);

    // 4) fused softmax + trilinear deformable sampling
    msda3d_sample<<<dim3(M_q), 256, 0, stream>>>(
        ws_value, ws_off, ws_attn, refp, ws_outp);

    // 5) output projection: (16384,256) @ (256,256) + b_out -> d_out
    gemm_f32_wmma<<<dim3(M_q / BM, DM / BN), 256, 0, stream>>>(
        ws_outp, W_out, b_out, out, M_q, DM, DM);
}